// InContextAssoMemBlock_14534169329964
// MI455X (gfx1250) — compile-verified
//
#include <hip/hip_runtime.h>
#include <hip/hip_bf16.h>
#include <math.h>

// Problem constants (from reference)
#define B_ 4
#define T_ 1024
#define C_ 768
#define NH_ 12
#define HS_ 64
#define EXP_SCALE_ 10.0f
#define KB_UP_ 5.0f
#define V_SHIFT_ 1

typedef __attribute__((ext_vector_type(2))) float v2f;
typedef __attribute__((ext_vector_type(8))) float v8f;

static __device__ __forceinline__ v8f wmma_f32(v2f a, v2f b, v8f c) {
  // V_WMMA_F32_16X16X4_F32 : D = A(16x4) * B(4x16) + C(16x16)
  return __builtin_amdgcn_wmma_f32_16x16x4_f32(false, a, false, b, (short)0, c,
                                               false, false);
}

// ---------------------------------------------------------------------------
// GEMM (NT): Out[m,n] = sum_k X[m,k] * W[n,k]   (einsum 'btc,dc->btd')
// One wave computes a 64x64 tile: 4x4 grid of 16x16 WMMA accumulators.
// Fragment layouts (CDNA5 ISA 7.12.2):
//   A 16x4 f32 : lane = (m%16) + 16*(k/2), vgpr = k%2
//   B 4x16 f32 : lane = n + 16*(k/2),      vgpr = k%2   (== A layout on W rows)
//   C/D 16x16  : vgpr = m%8, lane = n + 16*(m/8)
// ---------------------------------------------------------------------------
__global__ __launch_bounds__(256) void gemm_nt_64x64(
    const float* __restrict__ X, const float* __restrict__ W,
    float* __restrict__ Out, int M, int N, int K) {
  const int wave = blockIdx.x * 8 + (threadIdx.x >> 5);
  const int lane = threadIdx.x & 31;
  const int tilesN = N >> 6;
  const int nTiles = (M >> 6) * tilesN;
  if (wave >= nTiles) return;  // wave-uniform guard (EXEC stays all-ones)
  const int m0 = (wave / tilesN) << 6;
  const int n0 = (wave % tilesN) << 6;
  const int lr = lane & 15;  // row/col selector within 16
  const int lh = lane >> 4;  // selects k-pair {0,1} vs {2,3}

  v8f acc[4][4] = {};
  const float* Xp = X + (size_t)(m0 + lr) * K + 2 * lh;
  const float* Wp = W + (size_t)(n0 + lr) * K + 2 * lh;
  for (int k0 = 0; k0 < K; k0 += 4) {
    v2f a[4], b[4];
#pragma unroll
    for (int i = 0; i < 4; ++i) {
      a[i] = *(const v2f*)(Xp + (size_t)(i * 16) * K + k0);
      b[i] = *(const v2f*)(Wp + (size_t)(i * 16) * K + k0);
    }
#pragma unroll
    for (int i = 0; i < 4; ++i)
#pragma unroll
      for (int j = 0; j < 4; ++j) acc[i][j] = wmma_f32(a[i], b[j], acc[i][j]);
  }
#pragma unroll
  for (int i = 0; i < 4; ++i)
#pragma unroll
    for (int j = 0; j < 4; ++j)
#pragma unroll
      for (int r = 0; r < 8; ++r)
        Out[(size_t)(m0 + i * 16 + r + 8 * lh) * N + (n0 + j * 16 + lr)] =
            acc[i][j][r];
}

// ---------------------------------------------------------------------------
// EMA scan over time: decay @ k  ==  k[t] = e^{-λ} k[t-1] + (1-e^{-λ}) k_pre[t]
// In-place on the (B*T, C) buffer; one thread per (b, h, d).
// ---------------------------------------------------------------------------
__global__ __launch_bounds__(256) void ema_scan(float* __restrict__ kbuf,
                                                const float* __restrict__ lkb_in) {
  const int idx = blockIdx.x * blockDim.x + threadIdx.x;  // 0 .. B*NH*HS-1
  if (idx >= B_ * NH_ * HS_) return;
  const int d = idx % HS_;
  const int h = (idx / HS_) % NH_;
  const int b = idx / (HS_ * NH_);
  const float lkb = fabsf(lkb_in[h]) * EXP_SCALE_;
  const float e = expf(-lkb);
  const float w = 1.0f - e;
  float* p = kbuf + (size_t)b * T_ * C_ + h * HS_ + d;
  float s = 0.0f;
  for (int t = 0; t < T_; ++t) {
    float xv = p[(size_t)t * C_];
    s = e * s + w * xv;
    p[(size_t)t * C_] = s;
  }
}

// ---------------------------------------------------------------------------
// Normalize k rows (64-vec per head) and scale by exp(min(kernel_beta*10, 5)).
// One wave per (b*T + t, h); 2 elements per lane; wave32 shuffle reduction.
// ---------------------------------------------------------------------------
__global__ __launch_bounds__(256) void knorm(float* __restrict__ kbuf,
                                             const float* __restrict__ kernel_beta) {
  const int wave = blockIdx.x * 8 + (threadIdx.x >> 5);
  const int lane = threadIdx.x & 31;
  const int h = wave % NH_;
  const int bt = wave / NH_;
  if (bt >= B_ * T_) return;
  float* p = kbuf + (size_t)bt * C_ + h * HS_;
  float x0 = p[lane], x1 = p[lane + 32];
  float ss = x0 * x0 + x1 * x1;
#pragma unroll
  for (int m = 16; m >= 1; m >>= 1) ss += __shfl_xor(ss, m, 32);
  const float kb = expf(fminf(kernel_beta[h] * EXP_SCALE_, KB_UP_));
  const float sc = kb / sqrtf(ss);
  p[lane] = x0 * sc;
  p[lane + 32] = x1 * sc;
}

// ---------------------------------------------------------------------------
// v = (shift(lin)*(1-c) + lin*c), row-normalized, scaled by exp(value_beta*10)
// One wave per (b*T + t, h).
// ---------------------------------------------------------------------------
__global__ __launch_bounds__(256) void vbuild(const float* __restrict__ lin,
                                              float* __restrict__ vbuf,
                                              const float* __restrict__ coef,
                                              const float* __restrict__ value_beta) {
  const int wave = blockIdx.x * 8 + (threadIdx.x >> 5);
  const int lane = threadIdx.x & 31;
  const int h = wave % NH_;
  const int bt = wave / NH_;
  if (bt >= B_ * T_) return;
  const int t = bt % T_;
  const float* ph = lin + (size_t)bt * C_ + h * HS_;
  float vh0 = ph[lane], vh1 = ph[lane + 32];
  float vs0 = 0.0f, vs1 = 0.0f;
  if (t < T_ - 1) {  // shifted sample is next timestep within same batch
    vs0 = ph[C_ + lane];
    vs1 = ph[C_ + lane + 32];
  }
  const float c = coef[h];
  float v0 = vs0 * (1.0f - c) + vh0 * c;
  float v1 = vs1 * (1.0f - c) + vh1 * c;
  float ss = v0 * v0 + v1 * v1;
#pragma unroll
  for (int m = 16; m >= 1; m >>= 1) ss += __shfl_xor(ss, m, 32);
  const float sc = expf(value_beta[h] * EXP_SCALE_) / sqrtf(ss);
  float* po = vbuf + (size_t)bt * C_ + h * HS_;
  po[lane] = v0 * sc;
  po[lane + 32] = v1 * sc;
}

// ---------------------------------------------------------------------------
// Flash-style causal attention with shift-1 mask:
//   scores[i,j] = k[i]·k[j] for j <= i-1 ; row i==0 -> y = 0
//   y[i] = softmax_j(scores) @ v
// One wave handles 16 query rows. S and PV tiles via WMMA f32 16x16x4.
// P (C/D layout) -> A layout through per-wave LDS slab (intra-wave LDS ops
// are in-order on CDNA5; no barrier needed, waves have divergent trip counts).
// ---------------------------------------------------------------------------
__global__ __launch_bounds__(256) void attn_flash(
    const float* __restrict__ kmat, const float* __restrict__ vmat,
    float* __restrict__ ymat) {
  __shared__ __attribute__((aligned(16))) float lds[8][16 * 16];
  const int wslot = threadIdx.x >> 5;
  const int lane = threadIdx.x & 31;
  const int wave = blockIdx.x * 8 + wslot;
  const int nIT = T_ / 16;
  const int it = wave % nIT;
  const int bh = wave / nIT;
  if (bh >= B_ * NH_) return;
  const int h = bh % NH_;
  const int b = bh / NH_;
  const int i0 = it * 16;
  const int lr = lane & 15, lh = lane >> 4;

  const float* Kb = kmat + (size_t)b * T_ * C_ + h * HS_;  // (T, HS) stride C
  const float* Vb = vmat + (size_t)b * T_ * C_ + h * HS_;
  float* Yb = ymat + (size_t)b * T_ * C_ + h * HS_;

  // Q fragments: rows [i0, i0+16), K-dim HS=64 in 16 chunks of 4
  v2f qa[16];
#pragma unroll
  for (int kc = 0; kc < 16; ++kc)
    qa[kc] = *(const v2f*)(Kb + (size_t)(i0 + lr) * C_ + kc * 4 + 2 * lh);

  v8f acc[4] = {};  // y accumulator: 16 rows x 64 cols (4 N-subtiles)
  float mrow[8], lrow[8];
#pragma unroll
  for (int r = 0; r < 8; ++r) {
    mrow[r] = -INFINITY;
    lrow[r] = 0.0f;
  }

  for (int jt = 0; jt <= it; ++jt) {
    const int j0 = jt * 16;
    // S = Q @ K_j^T  (B-frag of K^T == A-style load of row-major K)
    v8f S = {};
#pragma unroll
    for (int kc = 0; kc < 16; ++kc) {
      v2f bf = *(const v2f*)(Kb + (size_t)(j0 + lr) * C_ + kc * 4 + 2 * lh);
      S = wmma_f32(qa[kc], bf, S);
    }
    // Causal shift-1 mask — only the diagonal tile is partially masked
    if (jt == it) {
#pragma unroll
      for (int r = 0; r < 8; ++r) {
        int m = r + 8 * lh;        // local row
        if (lr > m - V_SHIFT_) S[r] = -INFINITY;  // keep j <= i-1
      }
    }
    // Per-row tile max (reduce across the 16-lane group holding one row)
    float corr[8];
#pragma unroll
    for (int r = 0; r < 8; ++r) {
      float v = S[r];
#pragma unroll
      for (int msk = 1; msk < 16; msk <<= 1) v = fmaxf(v, __shfl_xor(v, msk, 32));
      float mn = fmaxf(mrow[r], v);
      corr[r] = (mrow[r] == -INFINITY) ? 0.0f : expf(mrow[r] - mn);
      mrow[r] = mn;
    }
    // P = exp(S - m); row sums; rescale running state
#pragma unroll
    for (int r = 0; r < 8; ++r) {
      float p = (S[r] == -INFINITY) ? 0.0f : expf(S[r] - mrow[r]);
      S[r] = p;
      float rs = p;
#pragma unroll
      for (int msk = 1; msk < 16; msk <<= 1) rs += __shfl_xor(rs, msk, 32);
      lrow[r] = lrow[r] * corr[r] + rs;
    }
#pragma unroll
    for (int nn = 0; nn < 4; ++nn)
#pragma unroll
      for (int r = 0; r < 8; ++r) acc[nn][r] *= corr[r];

    // Stage P (C layout) into LDS as row-major 16x16, reload as A fragments
    float* pl = lds[wslot];
#pragma unroll
    for (int r = 0; r < 8; ++r) pl[(r + 8 * lh) * 16 + lr] = S[r];
#pragma unroll
    for (int kc = 0; kc < 4; ++kc) {  // K=16 in 4 chunks of 4
      v2f pa = *(const v2f*)(pl + lr * 16 + kc * 4 + 2 * lh);
#pragma unroll
      for (int nn = 0; nn < 4; ++nn) {
        // B[k,n] = V[j0+k, nn*16+n]
        const float* vp = Vb + (size_t)(j0 + kc * 4 + 2 * lh) * C_ + nn * 16 + lr;
        v2f vb;
        vb.x = vp[0];
        vb.y = vp[C_];
        acc[nn] = wmma_f32(pa, vb, acc[nn]);
      }
    }
  }

  // Finalize: y = acc / l ; rows i < V_SHIFT forced to zero
#pragma unroll
  for (int r = 0; r < 8; ++r) {
    const int i = i0 + r + 8 * lh;
    const float inv = (lrow[r] > 0.0f) ? 1.0f / lrow[r] : 0.0f;
#pragma unroll
    for (int nn = 0; nn < 4; ++nn) {
      float yv = acc[nn][r] * inv;
      if (i < V_SHIFT_) yv = 0.0f;
      Yb[(size_t)i * C_ + nn * 16 + lr] = yv;
    }
  }
}

// ---------------------------------------------------------------------------
extern "C" void kernel_launch(void* const* d_in, const int* in_sizes, int n_in,
                              void* d_out, int out_size, void* d_ws,
                              size_t ws_size, hipStream_t stream) {
  const float* x = (const float*)d_in[0];
  const float* conv_w = (const float*)d_in[1];
  const float* lin_w = (const float*)d_in[2];
  const float* coef = (const float*)d_in[3];
  const float* leaky_key_beta = (const float*)d_in[4];
  const float* kernel_beta = (const float*)d_in[5];
  const float* value_beta = (const float*)d_in[6];
  const float* proj_w = (const float*)d_in[7];
  float* out = (float*)d_out;

  const size_t SZ = (size_t)B_ * T_ * C_ * sizeof(float);  // 12.58 MB
  float* kbuf = (float*)d_ws;                  // k_pre -> k (in place)
  float* linbuf = (float*)((char*)d_ws + SZ);  // lin, later reused for y
  float* vbuf = (float*)((char*)d_ws + 2 * SZ);
  float* ybuf = linbuf;  // lin is dead after vbuild

  const int M = B_ * T_, N = C_, K = C_;
  const int gemmBlocks = ((M >> 6) * (N >> 6) + 7) / 8;  // 96
  const int rowWaves = B_ * T_ * NH_;                    // 49152
  const int rowBlocks = (rowWaves + 7) / 8;              // 6144
  const int attnBlocks = (B_ * NH_ * (T_ / 16) + 7) / 8; // 384

  // 1) k_pre = x @ conv_w^T ; 2) lin = x @ lin_w^T
  gemm_nt_64x64<<<gemmBlocks, 256, 0, stream>>>(x, conv_w, kbuf, M, N, K);
  gemm_nt_64x64<<<gemmBlocks, 256, 0, stream>>>(x, lin_w, linbuf, M, N, K);
  // 3) decay matmul as EMA scan (in place)
  ema_scan<<<(B_ * NH_ * HS_ + 255) / 256, 256, 0, stream>>>(kbuf,
                                                             leaky_key_beta);
  // 4) normalize + scale k
  knorm<<<rowBlocks, 256, 0, stream>>>(kbuf, kernel_beta);
  // 5) build v (blend, normalize, scale)
  vbuild<<<rowBlocks, 256, 0, stream>>>(linbuf, vbuf, coef, value_beta);
  // 6) causal flash attention: y = softmax(k k^T, shifted-causal) @ v
  attn_flash<<<attnBlocks, 256, 0, stream>>>(kbuf, vbuf, ybuf);
  // 7) out = y @ proj_w^T
  gemm_nt_64x64<<<gemmBlocks, 256, 0, stream>>>(ybuf, proj_w, out, M, N, K);
}